// KeyPointSimpleField_42880953483729
// MI455X (gfx1250) — compile-verified
//
#include <hip/hip_runtime.h>
#include <hip/hip_bf16.h>
#include <math.h>

#define BB        2
#define NQ        4096
#define NS        16384
#define KNN       30
#define SEM_DIM   32
#define DESC_DIM  48
#define OUT_DIM   80
#define QPW       32              // queries per block (two 16x16 WMMA tiles)
#define CH        32              // scene candidates per chunk per wave
#define WAVES     4               // waves per block, scene-split
#define SPLIT     (NS / WAVES)    // 4096 candidates per wave
#define SLICE     (OUT_DIM / WAVES) // 20 output dims per wave in epilogue

typedef __attribute__((ext_vector_type(2))) float v2f;
typedef __attribute__((ext_vector_type(8))) float v8f;

__global__ __launch_bounds__(32 * WAVES)
void knn_field_kernel(const float* __restrict__ qpts,    // [B,NQ,3]
                      const float* __restrict__ spts,    // [B,NS,3]
                      const float* __restrict__ sdesc,   // [B,NS,48]
                      const float* __restrict__ semtab,  // [21,32]
                      const float* __restrict__ fpdesc,  // [48]
                      const int*   __restrict__ ssem,    // [B,NS]
                      float* __restrict__ out)           // [B,NQ,80]
{
    __shared__ float d2tile[WAVES][QPW][CH + 1];   // per-wave tile, stride 33
    __shared__ float mdist[WAVES][QPW][KNN];       // per-wave sorted top-K
    __shared__ int   midx [WAVES][QPW][KNN];

    const int lane = threadIdx.x & 31;
    const int wid  = threadIdx.x >> 5;
    const int blocksPerBatch = NQ / QPW;
    const int b  = blockIdx.x / blocksPerBatch;
    const int q0 = (blockIdx.x % blocksPerBatch) * QPW;
    const int q  = q0 + lane;                 // query owned by this lane

    const float* __restrict__ qb = qpts + (size_t)b * NQ * 3;
    const float* __restrict__ sb = spts + (size_t)b * NS * 3;

    // Owned query + |q|^2 (lane L holds q2 of query q0+L)
    const float qx = qb[q * 3 + 0], qy = qb[q * 3 + 1], qz = qb[q * 3 + 2];
    const float q2own = qx * qx + qy * qy + qz * qz;

    // --- A matrices (16x4, K padded 3->4), branchless construction --------
    // lanes 0-15 hold {x,y} (K=0,1); lanes 16-31 hold {z,0} (K=2,3)
    const int  am = lane & 15;
    const bool lo = (lane < 16);
    v2f a0, a1;
    {
        const float ax0 = qb[(q0 + am) * 3 + 0];
        const float ay0 = qb[(q0 + am) * 3 + 1];
        const float az0 = qb[(q0 + am) * 3 + 2];
        const float ax1 = qb[(q0 + 16 + am) * 3 + 0];
        const float ay1 = qb[(q0 + 16 + am) * 3 + 1];
        const float az1 = qb[(q0 + 16 + am) * 3 + 2];
        a0.x = lo ? ax0 : az0;  a0.y = lo ? ay0 : 0.0f;
        a1.x = lo ? ax1 : az1;  a1.y = lo ? ay1 : 0.0f;
    }

    // --- Loop-invariant q^2 row broadcasts (hoisted) -----------------------
    // C layout: VGPR v -> row M=v (lanes 0-15) / M=v+8 (lanes 16-31)
    float q2r0[8], q2r1[8];
#pragma unroll
    for (int v = 0; v < 8; ++v) {
        const int m = lo ? v : (v + 8);
        q2r0[v] = __shfl(q2own, m, 32);        // tile0 (queries q0..q0+15)
        q2r1[v] = __shfl(q2own, m + 16, 32);   // tile1 (queries q0+16..q0+31)
    }

    // Private top-K (sorted ascending), register resident
    float bd[KNN]; int bi[KNN];
#pragma unroll
    for (int k = 0; k < KNN; ++k) { bd[k] = 3.0e38f; bi[k] = 0; }

    // --- branchless chunk load: all lanes read x,y,z of their column ------
    // bm = lanes 0-15 {x,y} / lanes 16-31 {z,0}; also full |s|^2 locally
    const int sbase = wid * SPLIT;
    const int send  = sbase + SPLIT;

    auto load_chunk = [&](int s, v2f& bA, v2f& bB, float& s2A, float& s2B) {
        const float xA = sb[(s + am) * 3 + 0];
        const float yA = sb[(s + am) * 3 + 1];
        const float zA = sb[(s + am) * 3 + 2];
        const float xB = sb[(s + 16 + am) * 3 + 0];
        const float yB = sb[(s + 16 + am) * 3 + 1];
        const float zB = sb[(s + 16 + am) * 3 + 2];
        bA.x = lo ? xA : zA;  bA.y = lo ? yA : 0.0f;
        bB.x = lo ? xB : zB;  bB.y = lo ? yB : 0.0f;
        s2A = xA * xA + yA * yA + zA * zA;
        s2B = xB * xB + yB * yB + zB * zB;
    };

    v2f bmA, bmB; float s2A, s2B;
    load_chunk(sbase, bmA, bmB, s2A, s2B);

    for (int s0 = sbase; s0 < send; s0 += CH) {
        // --- 4 dot-product tiles on the matrix unit -----------------------
        v8f c00 = {}, c01 = {}, c10 = {}, c11 = {};
        c00 = __builtin_amdgcn_wmma_f32_16x16x4_f32(false, a0, false, bmA,
                                                    (short)0, c00, false, false);
        c10 = __builtin_amdgcn_wmma_f32_16x16x4_f32(false, a1, false, bmA,
                                                    (short)0, c10, false, false);
        c01 = __builtin_amdgcn_wmma_f32_16x16x4_f32(false, a0, false, bmB,
                                                    (short)0, c01, false, false);
        c11 = __builtin_amdgcn_wmma_f32_16x16x4_f32(false, a1, false, bmB,
                                                    (short)0, c11, false, false);

        // --- prefetch next chunk (uniform clamp, no EXEC divergence) ------
        v2f nA, nB; float n2A, n2B;
        int sn = s0 + CH; if (sn > send - CH) sn = send - CH;
        load_chunk(sn, nA, nB, n2A, n2B);

        // --- d2 = q2[M] + s2[N] - 2*qs, scatter to this wave's LDS tile ---
#pragma unroll
        for (int v = 0; v < 8; ++v) {
            const int m = lo ? v : (v + 8);
            d2tile[wid][m][am]           = q2r0[v] + s2A - 2.0f * c00[v];
            d2tile[wid][m][16 + am]      = q2r0[v] + s2B - 2.0f * c01[v];
            d2tile[wid][m + 16][am]      = q2r1[v] + s2A - 2.0f * c10[v];
            d2tile[wid][m + 16][16 + am] = q2r1[v] + s2B - 2.0f * c11[v];
        }
        __builtin_amdgcn_wave_barrier();   // wave-private tile; DS is in-order

        // --- selection: each lane scans its own query's 32 candidates -----
        float worst = bd[KNN - 1];
#pragma unroll 4
        for (int n = 0; n < CH; ++n) {
            const float d = d2tile[wid][lane][n];
            if (d < worst) {
                bd[KNN - 1] = d;
                bi[KNN - 1] = s0 + n;
#pragma unroll
                for (int k = KNN - 1; k > 0; --k) {
                    const bool  sw = bd[k] < bd[k - 1];
                    const float td = bd[k - 1];
                    const int   ti = bi[k - 1];
                    bd[k - 1] = sw ? bd[k] : td;  bi[k - 1] = sw ? bi[k] : ti;
                    bd[k]     = sw ? td : bd[k];  bi[k]     = sw ? ti : bi[k];
                }
                worst = bd[KNN - 1];
            }
        }
        __builtin_amdgcn_wave_barrier();   // WAR before tile rewrite

        bmA = nA; bmB = nB; s2A = n2A; s2B = n2B;
    }

    // --- publish per-wave sorted lists ------------------------------------
#pragma unroll
    for (int k = 0; k < KNN; ++k) {
        mdist[wid][lane][k] = bd[k];
        midx [wid][lane][k] = bi[k];
    }
    __syncthreads();

    // --- 4-way merge (redundant per wave) + sliced epilogue ---------------
    float acc[SLICE];
#pragma unroll
    for (int j = 0; j < SLICE; ++j) acc[j] = 0.0f;

    const int* __restrict__ semb = ssem + (size_t)b * NS;
    const int jbase = wid * SLICE;

    int p0 = 0, p1 = 0, p2 = 0, p3 = 0;
    for (int k = 0; k < KNN; ++k) {
        const float h0 = mdist[0][lane][p0];
        const float h1 = mdist[1][lane][p1];
        const float h2 = mdist[2][lane][p2];
        const float h3 = mdist[3][lane][p3];
        const int   i0 = midx[0][lane][p0];
        const int   i1 = midx[1][lane][p1];
        const int   i2 = midx[2][lane][p2];
        const int   i3 = midx[3][lane][p3];

        const float best = fminf(fminf(h0, h1), fminf(h2, h3));
        const bool  t0 = (best == h0);
        const bool  t1 = !t0 && (best == h1);
        const bool  t2 = !t0 && !t1 && (best == h2);
        const bool  t3 = !t0 && !t1 && !t2;
        const int   gi = t0 ? i0 : (t1 ? i1 : (t2 ? i2 : i3));
        p0 += t0; p1 += t1; p2 += t2; p3 += t3;

        const float d   = sqrtf(fmaxf(best, 1e-12f));
        const float wgt = __expf(-d) * (1.0f / (float)KNN);
        const int   sm  = semb[gi];
        const float* __restrict__ dsc =
            (sm < 0) ? fpdesc : (sdesc + ((size_t)b * NS + gi) * DESC_DIM);
        const float* __restrict__ se = semtab + (size_t)(sm + 1) * SEM_DIM;

#pragma unroll
        for (int jj = 0; jj < SLICE; ++jj) {
            const int j = jbase + jj;
            const float v = (j < DESC_DIM) ? dsc[j] : se[j - DESC_DIM];
            acc[jj] = fmaf(wgt, v, acc[jj]);
        }
    }

    float* __restrict__ op = out + ((size_t)b * NQ + q) * OUT_DIM + jbase;
#pragma unroll
    for (int jj = 0; jj < SLICE; ++jj) op[jj] = acc[jj];
}

extern "C" void kernel_launch(void* const* d_in, const int* in_sizes, int n_in,
                              void* d_out, int out_size, void* d_ws, size_t ws_size,
                              hipStream_t stream) {
    (void)in_sizes; (void)n_in; (void)d_ws; (void)ws_size; (void)out_size;
    const float* qpts   = (const float*)d_in[0];
    const float* spts   = (const float*)d_in[1];
    const float* sdesc  = (const float*)d_in[2];
    const float* semtab = (const float*)d_in[3];
    const float* fpdesc = (const float*)d_in[4];
    const int*   ssem   = (const int*)d_in[5];
    float* out = (float*)d_out;

    dim3 grid(BB * (NQ / QPW));      // 256 blocks
    dim3 block(32 * WAVES);          // 4 waves, scene-split
    knn_field_kernel<<<grid, block, 0, stream>>>(qpts, spts, sdesc, semtab,
                                                 fpdesc, ssem, out);
}